// GaussianLanguageModel_2482491097307
// MI455X (gfx1250) — compile-verified
//
#include <hip/hip_runtime.h>
#include <math.h>

#define DIMN 16
#define VOCAB 2000
#define TSTEPS 256
#define LOG2PI 1.8378770664093453f

typedef __attribute__((ext_vector_type(2))) float v2f;
typedef __attribute__((ext_vector_type(8))) float v8f;

// ---------------------------------------------------------------------------
// Kernel A: setup (lam_t, eta_t) + sequential Kalman scan. Single wave (32 ln).
// Gauss-Jordan inverses in LDS, one column per lane; single-wave lockstep makes
// the read-of-column-k-before-overwrite safe within each elimination row.
// ---------------------------------------------------------------------------
__global__ __launch_bounds__(32)
void glm_scan_kernel(const int* __restrict__ sent,
                     const float* __restrict__ emu,    // VOCAB x 16
                     const float* __restrict__ echo,   // VOCAB x 16
                     const float* __restrict__ tmu,    // 32
                     const float* __restrict__ tcho,   // 32 x 32
                     float* __restrict__ pred_mus,     // 256 x 16
                     float* __restrict__ pred_vars)    // 256 x 256
{
    __shared__ float LamT[32][33];   // lam_t
    __shared__ float Aug[32][66];    // 32x64 augmented GJ (+pad)
    __shared__ float A16[16][34];    // 16x32 augmented GJ (+pad)
    __shared__ float etaT[32];
    __shared__ float ev[32];
    __shared__ float mu[32];
    __shared__ float pmu[16];
    __shared__ float pvar[16][17];
    __shared__ float wmu[16];
    __shared__ float av[16];         // 1/word_var
    __shared__ float lmid[16][17];
    __shared__ float emid[16];

    const int lane = threadIdx.x;

    // trans_var = tcho^T @ tcho  (lane owns column `lane`), augmented with I
    for (int i = 0; i < 32; ++i) {
        float s = 0.f;
        for (int k = 0; k < 32; ++k) s += tcho[k * 32 + i] * tcho[k * 32 + lane];
        Aug[i][lane]      = s;
        Aug[i][32 + lane] = (i == lane) ? 1.f : 0.f;
    }
    __syncthreads();
    // Gauss-Jordan invert 32x32
    for (int k = 0; k < 32; ++k) {
        float piv = 1.f / Aug[k][k];
        __syncthreads();
        Aug[k][lane]      *= piv;
        Aug[k][32 + lane] *= piv;
        __syncthreads();
        for (int i = 0; i < 32; ++i) {
            if (i == k) continue;
            float f = Aug[i][k];                 // all lanes load before stores
            Aug[i][lane]      -= f * Aug[k][lane];
            Aug[i][32 + lane] -= f * Aug[k][32 + lane];
        }
        __syncthreads();
    }
    for (int i = 0; i < 32; ++i) LamT[i][lane] = Aug[i][32 + lane];
    __syncthreads();
    {   // eta_t = lam_t @ tmu
        float s = 0.f;
        for (int j = 0; j < 32; ++j) s += LamT[lane][j] * tmu[j];
        etaT[lane] = s;
    }
    __syncthreads();

    for (int t = 0; t < TSTEPS; ++t) {
        const int w = sent[t];
        if (lane < 16) {
            float c = echo[w * 16 + lane];
            av[lane]  = 1.f / (c * c);
            wmu[lane] = emu[w * 16 + lane];
        }
        __syncthreads();

        if (t == 0) {
            // integral(word_mu[0], diag(word_var[0])): lam_mid=diag(av), eta_mid=wmu*av
            if (lane < 16) {
                for (int j = 0; j < 16; ++j) lmid[lane][j] = 0.f;
                lmid[lane][lane] = av[lane];
                emid[lane] = wmu[lane] * av[lane];
            }
            __syncthreads();
        } else {
            // lam0 = inv(pvar) via 16x32 GJ (lane owns column `lane` of [P|I])
            for (int i = 0; i < 16; ++i)
                A16[i][lane] = (lane < 16) ? pvar[i][lane]
                                           : ((lane - 16 == i) ? 1.f : 0.f);
            __syncthreads();
            for (int k = 0; k < 16; ++k) {
                float piv = 1.f / A16[k][k];
                __syncthreads();
                A16[k][lane] *= piv;
                __syncthreads();
                for (int i = 0; i < 16; ++i) {
                    if (i == k) continue;
                    float f = A16[i][k];
                    A16[i][lane] -= f * A16[k][lane];
                }
                __syncthreads();
            }
            // lam_mid = lam0 + diag(av); eta_mid = lam0@pmu + wmu*av
            if (lane < 16) {
                for (int i = 0; i < 16; ++i)
                    lmid[i][lane] = A16[i][16 + lane] + ((i == lane) ? av[lane] : 0.f);
                float s = 0.f;
                for (int j = 0; j < 16; ++j) s += A16[lane][16 + j] * pmu[j];
                emid[lane] = s + wmu[lane] * av[lane];
            }
            __syncthreads();
        }

        // M = lam_t; M[:16,:16] += lam_mid; e = eta_t; e[:16] += eta_mid
        for (int i = 0; i < 32; ++i) {
            float addm = (i < 16 && lane < 16) ? lmid[i][lane] : 0.f;
            Aug[i][lane]      = LamT[i][lane] + addm;
            Aug[i][32 + lane] = (i == lane) ? 1.f : 0.f;
        }
        ev[lane] = etaT[lane] + ((lane < 16) ? emid[lane] : 0.f);
        __syncthreads();
        // sig = inv(M), 32x32 GJ
        for (int k = 0; k < 32; ++k) {
            float piv = 1.f / Aug[k][k];
            __syncthreads();
            Aug[k][lane]      *= piv;
            Aug[k][32 + lane] *= piv;
            __syncthreads();
            for (int i = 0; i < 32; ++i) {
                if (i == k) continue;
                float f = Aug[i][k];
                Aug[i][lane]      -= f * Aug[k][lane];
                Aug[i][32 + lane] -= f * Aug[k][32 + lane];
            }
            __syncthreads();
        }
        {   // mu = sig @ e
            float s = 0.f;
            for (int j = 0; j < 32; ++j) s += Aug[lane][32 + j] * ev[j];
            mu[lane] = s;
        }
        __syncthreads();
        // carry + outputs: pm = mu[16:], pv = sig[16:,16:]
        if (lane < 16) {
            pmu[lane] = mu[16 + lane];
            pred_mus[t * 16 + lane] = mu[16 + lane];
            for (int i = 0; i < 16; ++i) {
                float s = Aug[16 + i][48 + lane];
                pvar[i][lane] = s;
                pred_vars[t * 256 + i * 16 + lane] = s;
            }
        }
        __syncthreads();
    }
}

// ---------------------------------------------------------------------------
// Kernel B: per-t natural parameters. lam0 = inv(pvar) built as W^T W with
// W = chol(pvar)^{-1}, computed by 4 chained V_WMMA_F32_16X16X4_F32 (f32, exact
// 16x16x16 product). A-layout of W^T equals B-layout of W -> one reg pair.
// ---------------------------------------------------------------------------
__global__ __launch_bounds__(32)
void glm_pert_kernel(const float* __restrict__ pred_mus,
                     const float* __restrict__ pred_vars,
                     float* __restrict__ lam0_all,   // 256 x 256
                     float* __restrict__ eta0_all,   // 256 x 16
                     float* __restrict__ zeta0_all)  // 256
{
    const int t = blockIdx.x;
    const int lane = threadIdx.x;
    __shared__ float P[16][17];     // cholesky of pvar (in place, lower)
    __shared__ float W[16][17];     // inv(L), lower triangular
    __shared__ float Lam[16][17];   // lam0
    __shared__ float pm[16];
    __shared__ float e0[16];

    for (int i = lane; i < 256; i += 32) P[i >> 4][i & 15] = pred_vars[t * 256 + i];
    if (lane < 16) pm[lane] = pred_mus[t * 16 + lane];
    __syncthreads();

    // right-looking Cholesky of P
    for (int j = 0; j < 16; ++j) {
        if (lane == 0) P[j][j] = sqrtf(P[j][j]);
        __syncthreads();
        if (lane > j && lane < 16) P[lane][j] /= P[j][j];
        __syncthreads();
        if (lane > j && lane < 16) {
            float lij = P[lane][j];
            for (int k = j + 1; k <= lane; ++k) P[lane][k] -= lij * P[k][j];
        }
        __syncthreads();
    }
    // W = inv(L): lane j solves L w = e_j by forward substitution
    if (lane < 16) {
        const int j = lane;
        for (int i = 0; i < 16; ++i) {
            if (i < j) { W[i][j] = 0.f; continue; }
            float s = (i == j) ? 1.f : 0.f;
            for (int k = j; k < i; ++k) s -= P[i][k] * W[k][j];
            W[i][j] = s / P[i][i];
        }
    }
    __syncthreads();

    // lam0 = W^T @ W  via 4x v_wmma_f32_16x16x4_f32 (all 32 lanes, EXEC=all 1s)
    const int half = lane >> 4;
    const int lm   = lane & 15;
    v8f acc = {};
#pragma unroll
    for (int kb = 0; kb < 4; ++kb) {
        const int k0 = kb * 4 + half * 2;
        v2f ab;
        ab.x = W[k0][lm];
        ab.y = W[k0 + 1][lm];
        // D = A x B + C, A = W^T (16x4 chunk), B = W (4x16 chunk); same regs.
        acc = __builtin_amdgcn_wmma_f32_16x16x4_f32(
            false, ab, false, ab, (short)0, acc, false, false);
    }
    // C/D layout: VGPR r -> (M = r + 8*half, N = lm)
#pragma unroll
    for (int r = 0; r < 8; ++r) {
        const int M = r + 8 * half;
        Lam[M][lm] = acc[r];
        lam0_all[t * 256 + M * 16 + lm] = acc[r];
    }
    __syncthreads();

    if (lane < 16) {   // eta0 = lam0 @ pm
        float s = 0.f;
        for (int j = 0; j < 16; ++j) s += Lam[lane][j] * pm[j];
        e0[lane] = s;
        eta0_all[t * 16 + lane] = s;
    }
    __syncthreads();
    if (lane == 0) {
        float ldet0 = 0.f;                       // log det lam0 = -2 sum log diag(cholP)
        for (int i = 0; i < 16; ++i) ldet0 -= logf(P[i][i]);
        ldet0 *= 2.f;
        float dot = 0.f;
        for (int i = 0; i < 16; ++i) dot += e0[i] * pm[i];
        zeta0_all[t] = -0.5f * (16.f * LOG2PI - ldet0 + dot);
    }
}

// ---------------------------------------------------------------------------
// Kernel C: one thread per (t,v). Register-resident packed 16x16 Cholesky of
// lam0[t] + diag(lam1[v]); Q = ||L^{-1} eta_new||^2 (forward solve only).
// ---------------------------------------------------------------------------
__global__ __launch_bounds__(256)
void glm_pair_kernel(const float* __restrict__ dmu,
                     const float* __restrict__ dcho,
                     const float* __restrict__ lam0_all,
                     const float* __restrict__ eta0_all,
                     const float* __restrict__ zeta0_all,
                     float* __restrict__ logits)     // 256 x VOCAB
{
    const int t = blockIdx.y;
    const int v = blockIdx.x * 256 + threadIdx.x;
    __shared__ float lam0s[256];
    __shared__ float e0s[16];
    __shared__ float z0s;

    lam0s[threadIdx.x] = lam0_all[t * 256 + threadIdx.x];
    if (threadIdx.x < 16) e0s[threadIdx.x] = eta0_all[t * 16 + threadIdx.x];
    if (threadIdx.x == 0) z0s = zeta0_all[t];
    __syncthreads();
    if (v >= VOCAB) return;

    __builtin_prefetch(&dcho[v * 16], 0, 1);   // global_prefetch_b8
    __builtin_prefetch(&dmu[v * 16], 0, 1);

    float lam1[16], eta1[16];
    float zlog = 0.f, zdot = 0.f;
#pragma unroll
    for (int i = 0; i < 16; ++i) {
        float c = dcho[v * 16 + i];
        float m = dmu[v * 16 + i];
        float l = 1.f / (c * c);
        lam1[i] = l;
        eta1[i] = m * l;
        zlog += logf(l);
        zdot += eta1[i] * m;
    }
    const float part1 = 16.f * LOG2PI;
    const float zeta1 = -0.5f * (part1 - zlog + zdot);

    // packed lower triangle S[idx(i,j)] = lam0 + diag(lam1)
    float S[136];
#pragma unroll
    for (int i = 0; i < 16; ++i) {
#pragma unroll
        for (int j = 0; j <= i; ++j) S[i * (i + 1) / 2 + j] = lam0s[i * 16 + j];
        S[i * (i + 1) / 2 + i] += lam1[i];
    }
    // in-place Cholesky (fully unrolled -> stays in VGPRs)
    float ldet = 0.f;
#pragma unroll
    for (int j = 0; j < 16; ++j) {
        float d = S[j * (j + 1) / 2 + j];
#pragma unroll
        for (int k = 0; k < j; ++k) { float l = S[j * (j + 1) / 2 + k]; d -= l * l; }
        d = sqrtf(d);
        S[j * (j + 1) / 2 + j] = d;
        ldet += logf(d);
        const float inv = 1.f / d;
#pragma unroll
        for (int i2 = j + 1; i2 < 16; ++i2) {
            float s = S[i2 * (i2 + 1) / 2 + j];
#pragma unroll
            for (int k = 0; k < j; ++k)
                s -= S[i2 * (i2 + 1) / 2 + k] * S[j * (j + 1) / 2 + k];
            S[i2 * (i2 + 1) / 2 + j] = s * inv;
        }
    }
    const float ldet_new = 2.f * ldet;

    // forward solve L y = eta_new; Q = |y|^2
    float y[16];
    float Q = 0.f;
#pragma unroll
    for (int i = 0; i < 16; ++i) {
        float s = e0s[i] + eta1[i];
#pragma unroll
        for (int k = 0; k < i; ++k) s -= S[i * (i + 1) / 2 + k] * y[k];
        y[i] = s / S[i * (i + 1) / 2 + i];
        Q += y[i] * y[i];
    }
    const float zeta_new = -0.5f * (part1 - ldet_new + Q);
    logits[t * VOCAB + v] = z0s + zeta1 - zeta_new;
}

// ---------------------------------------------------------------------------
// Kernel D: per-t log-sum-exp + NLL for the target token.
// ---------------------------------------------------------------------------
__global__ __launch_bounds__(256)
void glm_loss_kernel(const float* __restrict__ logits,
                     const int* __restrict__ sent,
                     float* __restrict__ per_t)
{
    const int t = blockIdx.x;
    const int tid = threadIdx.x;
    __shared__ float red[256];

    float m = -INFINITY;
    for (int v = tid; v < VOCAB; v += 256) m = fmaxf(m, logits[t * VOCAB + v]);
    red[tid] = m;
    __syncthreads();
    for (int s = 128; s > 0; s >>= 1) {
        if (tid < s) red[tid] = fmaxf(red[tid], red[tid + s]);
        __syncthreads();
    }
    const float gmax = red[0];
    __syncthreads();

    float sum = 0.f;
    for (int v = tid; v < VOCAB; v += 256) sum += expf(logits[t * VOCAB + v] - gmax);
    red[tid] = sum;
    __syncthreads();
    for (int s = 128; s > 0; s >>= 1) {
        if (tid < s) red[tid] += red[tid + s];
        __syncthreads();
    }
    if (tid == 0) {
        const float lse = gmax + logf(red[0]);
        const int target = sent[t + 1];
        per_t[t] = -(logits[t * VOCAB + target] - lse);
    }
}

__global__ __launch_bounds__(256)
void glm_mean_kernel(const float* __restrict__ per_t, float* __restrict__ out)
{
    __shared__ float red[256];
    red[threadIdx.x] = per_t[threadIdx.x];
    __syncthreads();
    for (int s = 128; s > 0; s >>= 1) {
        if (threadIdx.x < s) red[threadIdx.x] += red[threadIdx.x + s];
        __syncthreads();
    }
    if (threadIdx.x == 0) out[0] = red[0] * (1.f / 256.f);
}

// ---------------------------------------------------------------------------
extern "C" void kernel_launch(void* const* d_in, const int* in_sizes, int n_in,
                              void* d_out, int out_size, void* d_ws, size_t ws_size,
                              hipStream_t stream) {
    const int*   sent  = (const int*)  d_in[0];   // 257
    const float* emu   = (const float*)d_in[1];   // 2000x16
    const float* echo  = (const float*)d_in[2];   // 2000x16
    const float* tmu   = (const float*)d_in[3];   // 32
    const float* tcho  = (const float*)d_in[4];   // 32x32
    const float* dmu   = (const float*)d_in[5];   // 2000x16
    const float* dcho  = (const float*)d_in[6];   // 2000x16
    float* out = (float*)d_out;

    float* ws        = (float*)d_ws;
    float* pred_mus  = ws;                    // 256*16   = 4096
    float* pred_vars = pred_mus + 4096;       // 256*256  = 65536
    float* lam0_all  = pred_vars + 65536;     // 256*256  = 65536
    float* eta0_all  = lam0_all + 65536;      // 256*16   = 4096
    float* zeta0_all = eta0_all + 4096;       // 256
    float* logits    = zeta0_all + 256;       // 256*2000 = 512000
    float* per_t     = logits + 512000;       // 256

    glm_scan_kernel<<<1, 32, 0, stream>>>(sent, emu, echo, tmu, tcho,
                                          pred_mus, pred_vars);
    glm_pert_kernel<<<TSTEPS, 32, 0, stream>>>(pred_mus, pred_vars,
                                               lam0_all, eta0_all, zeta0_all);
    dim3 gridC((VOCAB + 255) / 256, TSTEPS);
    glm_pair_kernel<<<gridC, 256, 0, stream>>>(dmu, dcho, lam0_all, eta0_all,
                                               zeta0_all, logits);
    glm_loss_kernel<<<TSTEPS, 256, 0, stream>>>(logits, sent, per_t);
    glm_mean_kernel<<<1, 256, 0, stream>>>(per_t, out);
}